// CodeBook_52072183497474
// MI455X (gfx1250) — compile-verified
//
#include <hip/hip_runtime.h>

typedef __attribute__((ext_vector_type(2))) float v2f;
typedef __attribute__((ext_vector_type(8))) float v8f;

namespace {
constexpr int kN = 8192;
constexpr int kS = 32;
constexpr int kD = 32;
constexpr int kK = 1024;
constexpr float kBeta = 0.001f;
constexpr int kWavesPerBlock = 8;
constexpr int kBlockSize = kWavesPerBlock * 32;   // 256 threads = 8 wave32
constexpr int kTilesN = kN / 16;                  // 512 row tiles
constexpr int kWaves = kTilesN * kS;              // 16384 (tile, s) pairs
constexpr int kBlocks = kWaves / kWavesPerBlock;  // 2048 blocks
constexpr int kPairs = kK / 32;                   // 32 pairs of 16-code tiles
}

// ---- Kernel 1: per-code squared norms  wnorm[s*K+k] = sum_d W[s][k][d]^2
__global__ void vq_wnorm_kernel(const float* __restrict__ W, float* __restrict__ wnorm) {
  int i = blockIdx.x * blockDim.x + threadIdx.x;
  if (i < kS * kK) {
    const float* w = W + (size_t)i * kD;
    float acc = 0.f;
#pragma unroll
    for (int d = 0; d < kD; ++d) acc += w[d] * w[d];
    wnorm[i] = acc;
  }
}

// ---- Kernel 2: WMMA distance GEMM + argmin + gather + partial loss
__global__ __launch_bounds__(kBlockSize)
void vq_main_kernel(const float* __restrict__ z, const float* __restrict__ W,
                    const float* __restrict__ wnorm, float* __restrict__ out,
                    float* __restrict__ blockPartial) {
  __shared__ float wavePartial[kWavesPerBlock];
  const int lane  = threadIdx.x & 31;
  const int wave  = threadIdx.x >> 5;
  const int gwave = blockIdx.x * kWavesPerBlock + wave;
  const int s  = gwave & (kS - 1);        // codebook slice
  const int n0 = (gwave >> 5) * 16;       // first row of this wave's 16-row tile

  const int row = lane & 15;              // A row / B col / C col within tile
  const int hi  = lane >> 4;              // half-wave select

  // A fragments: zc tile 16x32, f32 WMMA 16x4 layout.
  // VGPR0: lanes0-15 -> K=4c+0, lanes16-31 -> K=4c+2; VGPR1: +1.
  // Register-resident across all 64 K-tiles.
  const float* zrow = z + (size_t)(n0 + row) * (kS * kD) + s * kD;
  v2f a[8];
#pragma unroll
  for (int c = 0; c < 8; ++c)
    a[c] = *(const v2f*)(zrow + 4 * c + 2 * hi);

  const float* Ws   = W + (size_t)s * kK * kD;
  const float* wn_s = wnorm + (size_t)s * kK;

  float bestVal[8];
  int   bestIdx[8];
#pragma unroll
  for (int j = 0; j < 8; ++j) { bestVal[j] = 3.402823466e+38f; bestIdx[j] = 0; }

  // Load both 16-code tiles of pair P: 16x b64 in one clause + 2 wnorm b32.
  auto loadPair = [&](int P, v2f (&b0)[8], v2f (&b1)[8], float& wn0, float& wn1) {
    const float* wk0 = Ws + (size_t)(P * 32 + row) * kD + 2 * hi;
    const float* wk1 = wk0 + 16 * kD;
#pragma unroll
    for (int c = 0; c < 8; ++c) {
      b0[c] = *(const v2f*)(wk0 + 4 * c);
      b1[c] = *(const v2f*)(wk1 + 4 * c);
    }
    wn0 = wn_s[P * 32 + row];
    wn1 = wn_s[P * 32 + 16 + row];
  };

  // Two independent WMMA accumulator chains, interleaved -> back-to-back
  // v_wmma instructions are independent (hides XDL accumulate latency).
  auto computePair = [&](const v2f (&b0)[8], const v2f (&b1)[8],
                         float wn0, float wn1, int kbase) {
    v8f acc0 = {};
    v8f acc1 = {};
#pragma unroll
    for (int c = 0; c < 8; ++c) {
      acc0 = __builtin_amdgcn_wmma_f32_16x16x4_f32(false, a[c], false, b0[c],
                                                   (short)0, acc0, false, false);
      acc1 = __builtin_amdgcn_wmma_f32_16x16x4_f32(false, a[c], false, b1[c],
                                                   (short)0, acc1, false, false);
    }
    // d2 = |zc|^2 + |W|^2 - 2G ; row-constant |zc|^2 dropped for argmin.
#pragma unroll
    for (int j = 0; j < 8; ++j) {
      float s0 = __builtin_fmaf(-2.0f, acc0[j], wn0);
      if (s0 < bestVal[j]) { bestVal[j] = s0; bestIdx[j] = kbase + row; }
      float s1 = __builtin_fmaf(-2.0f, acc1[j], wn1);
      if (s1 < bestVal[j]) { bestVal[j] = s1; bestIdx[j] = kbase + 16 + row; }
    }
  };

  // Software pipeline: prefetch pair P+1 while computing pair P.
  v2f bA0[8], bA1[8], bB0[8], bB1[8];
  float wnA0, wnA1, wnB0, wnB1;
  loadPair(0, bA0, bA1, wnA0, wnA1);
  for (int P = 0; P < kPairs; P += 2) {
    loadPair(P + 1, bB0, bB1, wnB0, wnB1);
    computePair(bA0, bA1, wnA0, wnA1, P * 32);
    if (P + 2 < kPairs)
      loadPair(P + 2, bA0, bA1, wnA0, wnA1);
    computePair(bB0, bB1, wnB0, wnB1, (P + 1) * 32);
  }

  // min-reduce over the 16 lanes holding the same C rows (xor stays in-half).
  // Tie-break to lowest index = jnp.argmin semantics.
#pragma unroll
  for (int j = 0; j < 8; ++j) {
    float v = bestVal[j];
    int  bi = bestIdx[j];
#pragma unroll
    for (int m = 1; m < 16; m <<= 1) {
      float ov = __shfl_xor(v, m, 32);
      int   oi = __shfl_xor(bi, m, 32);
      if (ov < v || (ov == v && oi < bi)) { v = ov; bi = oi; }
    }
    bestVal[j] = v; bestIdx[j] = bi;
  }
  // lanes0-15 now hold winners for rows 0..7; lanes16-31 for rows 8..15.

  float sq = 0.f;
  for (int r = 0; r < 16; ++r) {
    int idx = __shfl(bestIdx[r & 7], (r < 8) ? 0 : 16, 32);
    size_t zoff = (size_t)(n0 + r) * (kS * kD) + s * kD + lane;  // lane = d
    float wv = Ws[(size_t)idx * kD + lane];
    float zv = z[zoff];
    out[zoff] = wv;                 // zq_st forward value == zq
    float diff = wv - zv;
    sq += diff * diff;
  }

  // fixed-order reductions => deterministic loss (no float atomics)
#pragma unroll
  for (int m = 1; m < 32; m <<= 1) sq += __shfl_xor(sq, m, 32);
  if (lane == 0) wavePartial[wave] = sq;
  __syncthreads();
  if (threadIdx.x == 0) {
    float bsum = 0.f;
    for (int w = 0; w < kWavesPerBlock; ++w) bsum += wavePartial[w];
    blockPartial[blockIdx.x] = bsum;
  }
}

// ---- Kernel 3: final loss = S*(1+beta)*sum_sq / (N*S*D), fixed order
__global__ void vq_loss_kernel(const float* __restrict__ blockPartial,
                               float* __restrict__ lossOut) {
  if (threadIdx.x == 0 && blockIdx.x == 0) {
    double acc = 0.0;
    for (int i = 0; i < kBlocks; ++i) acc += (double)blockPartial[i];
    double mean = acc / ((double)kN * (double)kS * (double)kD);
    lossOut[0] = (float)((double)kS * (1.0 + (double)kBeta) * mean);
  }
}

extern "C" void kernel_launch(void* const* d_in, const int* in_sizes, int n_in,
                              void* d_out, int out_size, void* d_ws, size_t ws_size,
                              hipStream_t stream) {
  (void)in_sizes; (void)n_in; (void)out_size; (void)ws_size;
  const float* z = (const float*)d_in[0];
  const float* W = (const float*)d_in[1];
  float* out     = (float*)d_out;
  float* wnorm   = (float*)d_ws;            // kS*kK floats   (128 KB)
  float* partial = wnorm + kS * kK;         // kBlocks floats (8 KB)
  float* lossOut = out + (size_t)kN * kS * kD;

  vq_wnorm_kernel<<<(kS * kK + 255) / 256, 256, 0, stream>>>(W, wnorm);
  vq_main_kernel<<<kBlocks, kBlockSize, 0, stream>>>(z, W, wnorm, out, partial);
  vq_loss_kernel<<<1, 32, 0, stream>>>(partial, lossOut);
}